// LlamaAttentionMod_833223655927
// MI455X (gfx1250) — compile-verified
//
#include <hip/hip_runtime.h>
#include <hip/hip_bf16.h>
#include <math.h>

typedef __attribute__((ext_vector_type(2))) float v2f;
typedef __attribute__((ext_vector_type(8))) float v8f;

#define W_WORKERS 2
#define Q_LEN     16
#define HIDDEN    4096
#define N_HEADS   32
#define N_KV      8
#define HEAD_DIM  128
#define L_SHARED  8192
#define L_WORKER  2048

#define NT_SEG0   512            // 8192/16
#define NT_SEG1   129            // 2064/16
#define NT_TOTAL  770            // 512+129+129
#define CHUNKS    16             // split-K factor per (worker, head)
#define PART_STRIDE 2080         // 16 m + 16 l + 16*128 acc floats

// workspace offsets (in floats)
#define WS_Q    ((size_t)0)          // 2*32*16*128 = 131072
#define WS_KRAW ((size_t)131072)     // 2*8*16*128  =  32768
#define WS_KROT ((size_t)163840)     //               32768
#define WS_V    ((size_t)196608)     //               32768
#define WS_ATTN ((size_t)229376)     // 2*16*4096   = 131072
#define WS_PART ((size_t)360448)     // 64*16*2080  = 2129920

__device__ __forceinline__ v8f wmma4(v2f a, v2f b, v8f c) {
  // D = A(16x4,f32) * B(4x16,f32) + C(16x16,f32)
  return __builtin_amdgcn_wmma_f32_16x16x4_f32(false, a, false, b, (short)0, c,
                                               false, false);
}

// Async-copy one contiguous 8KB tile (16 rows x 128 f32) global -> LDS.
// Each lane moves 16B per instruction; INST_OFFSET advances both LDS and
// global addresses (ISA 15.18.3), so the copy is 16 unrolled b128 ops.
__device__ __forceinline__ void async_tile8k(const float* g, float* l, int lane) {
  unsigned    lds = (unsigned)(uintptr_t)l + lane * 16;
  const char* ga  = (const char*)g + lane * 16;
  #pragma unroll
  for (int i = 0; i < 16; ++i)
    asm volatile("global_load_async_to_lds_b128 %0, %1, off offset:%2"
                 :: "v"(lds), "v"(ga), "i"(i * 512) : "memory");
}

__device__ __forceinline__ void wait_async_keep32() {
  asm volatile("s_wait_asynccnt 0x20" ::: "memory");
}
__device__ __forceinline__ void wait_async_zero() {
  asm volatile("s_wait_asynccnt 0x0" ::: "memory");
}

// ---------------------------------------------------------------------------
// Kernel 1: fused QKV projection. One wave computes one 16x16 output tile of
// the M=16(tokens), N=6144(q|k|v), K=4096 GEMM (per worker).
// ---------------------------------------------------------------------------
__global__ __launch_bounds__(32) void qkv_gemm(
    const float* __restrict__ hs,
    const float* __restrict__ Wq, const float* __restrict__ bq,
    const float* __restrict__ Wk, const float* __restrict__ bk,
    const float* __restrict__ Wv, const float* __restrict__ bv,
    float* __restrict__ ws) {
  const int lane = threadIdx.x;
  const int w    = blockIdx.x / 384;
  const int n0   = (blockIdx.x % 384) * 16;

  const float* Wmat; const float* bias; int nloc;
  if (n0 < 4096)      { Wmat = Wq; bias = bq; nloc = n0; }
  else if (n0 < 5120) { Wmat = Wk; bias = bk; nloc = n0 - 4096; }
  else                { Wmat = Wv; bias = bv; nloc = n0 - 5120; }

  const int am  = lane & 15;             // A row (M) / B col (N)
  const int off = (lane >> 4) << 1;      // K sub-offset (0 or 2)
  const float* arow = hs   + (size_t)(w * Q_LEN + am) * HIDDEN + off;
  const float* brow = Wmat + (size_t)(nloc + am)      * HIDDEN + off;

  v8f acc = 0.0f;
  for (int k0 = 0; k0 < HIDDEN; k0 += 4) {
    __builtin_prefetch(brow + k0 + 1024, 0, 0);
    v2f a = *(const v2f*)(arow + k0);
    v2f b = *(const v2f*)(brow + k0);
    acc = wmma4(a, b, acc);
  }

  const float bval = bias[nloc + am];
  const int   mC   = (lane >> 4) << 3;
  const int   col  = n0 + am;
  #pragma unroll
  for (int slot = 0; slot < 8; ++slot) {
    const int mr  = slot + mC;
    const float v = acc[slot] + bval;
    if (col < 4096) {
      ws[WS_Q    + (((size_t)w * N_HEADS + (col >> 7)) * Q_LEN + mr) * HEAD_DIM + (col & 127)] = v;
    } else if (col < 5120) {
      const int j = col - 4096;
      ws[WS_KRAW + (((size_t)w * N_KV + (j >> 7)) * Q_LEN + mr) * HEAD_DIM + (j & 127)] = v;
    } else {
      const int j = col - 5120;
      ws[WS_V    + (((size_t)w * N_KV + (j >> 7)) * Q_LEN + mr) * HEAD_DIM + (j & 127)] = v;
    }
  }
}

// ---------------------------------------------------------------------------
// Kernel 2: RoPE on projected K (host-provided cos/sin).
// ---------------------------------------------------------------------------
__global__ void rope_k(const float* __restrict__ cosn,
                       const float* __restrict__ sinn,
                       float* __restrict__ ws) {
  const int idx = blockIdx.x * blockDim.x + threadIdx.x;
  if (idx >= W_WORKERS * N_KV * Q_LEN * HEAD_DIM) return;
  const int d = idx & 127;
  const int m = (idx >> 7) & 15;
  const int w = idx >> 14;
  const float* kr = ws + WS_KRAW;
  const float x  = kr[idx];
  const float xp = (d < 64) ? -kr[idx + 64] : kr[idx - 64];
  const float c  = cosn[(w * Q_LEN + m) * HEAD_DIM + d];
  const float s  = sinn[(w * Q_LEN + m) * HEAD_DIM + d];
  ws[WS_KROT + idx] = x * c + xp * s;
}

// ---------------------------------------------------------------------------
// Kernel 3: split-K flash attention partials. One wave per (w, h, chunk).
// Tiles are globally indexed 0..769: [0,512) seg0, [512,641) seg1, [641,770)
// seg2. K/V tiles are double-buffered in LDS via async global->LDS copies.
// ---------------------------------------------------------------------------
__global__ __launch_bounds__(32) void attn_partial(
    const float* __restrict__ ssk, const float* __restrict__ ssv,
    const float* __restrict__ w0k, const float* __restrict__ w0v,
    const float* __restrict__ w1k, const float* __restrict__ w1v,
    float* __restrict__ ws) {
  __shared__ float lds_rq[Q_LEN * HEAD_DIM];     // rotated Q for current seg
  __shared__ float lds_p[Q_LEN * 16];            // probability tile
  __shared__ float lds_k[2][Q_LEN * HEAD_DIM];   // double-buffered K tile
  __shared__ float lds_v[2][Q_LEN * HEAD_DIM];   // double-buffered V tile

  const int lane = threadIdx.x;
  const int c    = blockIdx.x & (CHUNKS - 1);
  const int hw   = blockIdx.x / CHUNKS;
  const int w    = hw >> 5;
  const int h    = hw & 31;
  const int kvh  = h >> 2;
  const float scale = 0.08838834764831845f;      // 128^-0.5

  const int mC  = (lane >> 4) << 3;
  const int nC  = lane & 15;
  const int off = (lane >> 4) << 1;

  const float* qbase = ws + WS_Q + (size_t)(w * N_HEADS + h) * Q_LEN * HEAD_DIM;
  const float* krot  = ws + WS_KROT;
  const float* vnew  = ws + WS_V;

  struct TileInfo { const float* K; const float* V; int base, r0, seg; };
  auto tinfo = [&](int g) {
    TileInfo ti;
    if (g < NT_SEG0) {
      ti.seg = 0; ti.r0 = g << 4; ti.base = 12304;
      ti.K = ssk + ((size_t)kvh * L_SHARED + ti.r0) * HEAD_DIM;
      ti.V = ssv + ((size_t)kvh * L_SHARED + ti.r0) * HEAD_DIM;
    } else {
      const int seg  = (g < NT_SEG0 + NT_SEG1) ? 1 : 2;
      const int t    = g - (seg == 1 ? NT_SEG0 : NT_SEG0 + NT_SEG1);
      const int wseg = seg - 1;
      ti.seg = seg; ti.r0 = t << 4;
      ti.base = (w == wseg) ? 2048 : 4112;
      if (ti.r0 < L_WORKER) {
        const float* wk = (seg == 1) ? w0k : w1k;
        const float* wv = (seg == 1) ? w0v : w1v;
        ti.K = wk + ((size_t)kvh * L_WORKER + ti.r0) * HEAD_DIM;
        ti.V = wv + ((size_t)kvh * L_WORKER + ti.r0) * HEAD_DIM;
      } else {
        ti.K = krot + ((size_t)(wseg * N_KV + kvh) * Q_LEN + (ti.r0 - L_WORKER)) * HEAD_DIM;
        ti.V = vnew + ((size_t)(wseg * N_KV + kvh) * Q_LEN + (ti.r0 - L_WORKER)) * HEAD_DIM;
      }
    }
    return ti;
  };

  const int gs = (c * NT_TOTAL) / CHUNKS;
  const int ge = ((c + 1) * NT_TOTAL) / CHUNKS;

  v8f acc[8];
  #pragma unroll
  for (int i = 0; i < 8; ++i) acc[i] = 0.0f;
  float mrow[8], lrow[8];
  #pragma unroll
  for (int i = 0; i < 8; ++i) { mrow[i] = -3.0e38f; lrow[i] = 0.0f; }

  int cur_seg = -1;

  // prologue: async-stage the first tile
  {
    TileInfo t0 = tinfo(gs);
    async_tile8k(t0.K, lds_k[0], lane);
    async_tile8k(t0.V, lds_v[0], lane);
  }

  for (int g = gs; g < ge; ++g) {
    const int buf = (g - gs) & 1;
    const TileInfo ti = tinfo(g);

    // stage next tile while this one is consumed
    if (g + 1 < ge) {
      TileInfo tn = tinfo(g + 1);
      async_tile8k(tn.K, lds_k[buf ^ 1], lane);
      async_tile8k(tn.V, lds_v[buf ^ 1], lane);
      wait_async_keep32();        // in-order: <=32 outstanding => tile g landed
    } else {
      wait_async_zero();
    }

    // (re)build segment-rotated Q in LDS when the segment changes
    if (ti.seg != cur_seg) {
      cur_seg = ti.seg;
      for (int idx = lane; idx < Q_LEN * HEAD_DIM; idx += 32) {
        const int mm = idx >> 7, d = idx & 127;
        const int i2 = (d & 63) * 2;
        const float inv = __expf(-(float)i2 * (9.210340371976184f / 128.0f));
        const float ang = (float)(ti.base + mm) * inv;
        float cc, s2; sincosf(ang, &s2, &cc);
        const float qv = qbase[idx];
        const float qp = (d < 64) ? -qbase[mm * HEAD_DIM + d + 64]
                                  :  qbase[mm * HEAD_DIM + d - 64];
        lds_rq[idx] = qv * cc + qp * s2;
      }
    }

    const float* kb = lds_k[buf];
    const float* vb = lds_v[buf];

    // ---- S = rq @ K^T (16x16 tile, K=128 in steps of 4) ----
    v8f s = 0.0f;
    const float* aK = lds_rq + (size_t)nC * HEAD_DIM + off;
    const float* bK = kb     + (size_t)nC * HEAD_DIM + off;
    for (int k0 = 0; k0 < HEAD_DIM; k0 += 4) {
      v2f a = *(const v2f*)(aK + k0);
      v2f b = *(const v2f*)(bK + k0);
      s = wmma4(a, b, s);
    }

    // ---- mask + online softmax ----
    float p[8];
    #pragma unroll
    for (int slot = 0; slot < 8; ++slot) {
      const int mr = slot + mC;
      const int r  = ti.r0 + nC;
      float val = s[slot] * scale;
      if (r > ti.base + mr) val = -3.0e38f;
      float rmax = val;
      rmax = fmaxf(rmax, __shfl_xor(rmax, 1, 16));
      rmax = fmaxf(rmax, __shfl_xor(rmax, 2, 16));
      rmax = fmaxf(rmax, __shfl_xor(rmax, 4, 16));
      rmax = fmaxf(rmax, __shfl_xor(rmax, 8, 16));
      const float mnew = fmaxf(mrow[slot], rmax);
      const float corr = __expf(mrow[slot] - mnew);
      const float pv   = __expf(val - mnew);
      float rsum = pv;
      rsum += __shfl_xor(rsum, 1, 16);
      rsum += __shfl_xor(rsum, 2, 16);
      rsum += __shfl_xor(rsum, 4, 16);
      rsum += __shfl_xor(rsum, 8, 16);
      lrow[slot] = lrow[slot] * corr + rsum;
      mrow[slot] = mnew;
      p[slot] = pv;
      #pragma unroll
      for (int nt = 0; nt < 8; ++nt) acc[nt][slot] *= corr;
    }

    // ---- stage P through LDS to re-fragment for PV ----
    #pragma unroll
    for (int slot = 0; slot < 8; ++slot)
      lds_p[(slot + mC) * 16 + nC] = p[slot];

    // ---- O += P @ V ----
    const float* aP = lds_p + (size_t)nC * 16 + off;
    #pragma unroll
    for (int nt = 0; nt < 8; ++nt) {
      const float* bV = vb + (size_t)off * HEAD_DIM + nt * 16 + nC;
      v8f o = acc[nt];
      #pragma unroll
      for (int kk = 0; kk < 16; kk += 4) {
        v2f a = *(const v2f*)(aP + kk);
        v2f b;
        b.x = bV[(size_t)kk * HEAD_DIM];
        b.y = bV[(size_t)kk * HEAD_DIM + HEAD_DIM];
        o = wmma4(a, b, o);
      }
      acc[nt] = o;
    }
  }

  // write un-normalized partial (m, l, acc)
  float* P = ws + WS_PART + (size_t)(hw * CHUNKS + c) * PART_STRIDE;
  if (nC == 0) {
    #pragma unroll
    for (int slot = 0; slot < 8; ++slot) {
      P[slot + mC]      = mrow[slot];
      P[16 + slot + mC] = lrow[slot];
    }
  }
  #pragma unroll
  for (int slot = 0; slot < 8; ++slot) {
    const int mr = slot + mC;
    #pragma unroll
    for (int nt = 0; nt < 8; ++nt)
      P[32 + (size_t)mr * HEAD_DIM + nt * 16 + nC] = acc[nt][slot];
  }
}

// ---------------------------------------------------------------------------
// Kernel 3b: merge the CHUNKS partials per (w,h) and normalize.
// ---------------------------------------------------------------------------
__global__ __launch_bounds__(256) void attn_combine(float* __restrict__ ws) {
  const int w = blockIdx.x >> 5;
  const int h = blockIdx.x & 31;
  __shared__ float wgt[CHUNKS][Q_LEN];
  __shared__ float Lrow[Q_LEN];
  const float* P0 = ws + WS_PART + (size_t)((w * N_HEADS + h) * CHUNKS) * PART_STRIDE;
  const int tid = threadIdx.x;

  if (tid < Q_LEN) {
    float M = -3.0e38f;
    for (int c = 0; c < CHUNKS; ++c)
      M = fmaxf(M, P0[(size_t)c * PART_STRIDE + tid]);
    float L = 0.0f;
    for (int c = 0; c < CHUNKS; ++c) {
      const float e = __expf(P0[(size_t)c * PART_STRIDE + tid] - M);
      wgt[c][tid] = e;
      L += e * P0[(size_t)c * PART_STRIDE + 16 + tid];
    }
    Lrow[tid] = L;
  }
  __syncthreads();

  float* outp = ws + WS_ATTN + (size_t)w * Q_LEN * (N_HEADS * HEAD_DIM) + h * HEAD_DIM;
  for (int idx = tid; idx < Q_LEN * HEAD_DIM; idx += 256) {
    const int r = idx >> 7, d = idx & 127;
    float sum = 0.0f;
    for (int c = 0; c < CHUNKS; ++c)
      sum += P0[(size_t)c * PART_STRIDE + 32 + (size_t)r * HEAD_DIM + d] * wgt[c][r];
    outp[(size_t)r * (N_HEADS * HEAD_DIM) + d] = sum / Lrow[r];
  }
}

// ---------------------------------------------------------------------------
// Kernel 4: output projection, M=16, N=4096, K=4096 per worker.
// ---------------------------------------------------------------------------
__global__ __launch_bounds__(32) void out_gemm(const float* __restrict__ Wo,
                                               const float* __restrict__ attn,
                                               float* __restrict__ out) {
  const int lane = threadIdx.x;
  const int w    = blockIdx.x / 256;
  const int n0   = (blockIdx.x % 256) * 16;
  const int am   = lane & 15;
  const int off  = (lane >> 4) << 1;
  const float* arow = attn + (size_t)(w * Q_LEN + am) * 4096 + off;
  const float* brow = Wo   + (size_t)(n0 + am)        * 4096 + off;

  v8f acc = 0.0f;
  for (int k0 = 0; k0 < 4096; k0 += 4) {
    __builtin_prefetch(brow + k0 + 1024, 0, 0);
    v2f a = *(const v2f*)(arow + k0);
    v2f b = *(const v2f*)(brow + k0);
    acc = wmma4(a, b, acc);
  }
  const int mC = (lane >> 4) << 3;
  #pragma unroll
  for (int slot = 0; slot < 8; ++slot)
    out[(size_t)(w * Q_LEN + slot + mC) * 4096 + n0 + am] = acc[slot];
}

// ---------------------------------------------------------------------------
extern "C" void kernel_launch(void* const* d_in, const int* in_sizes, int n_in,
                              void* d_out, int out_size, void* d_ws, size_t ws_size,
                              hipStream_t stream) {
  (void)in_sizes; (void)n_in; (void)out_size; (void)ws_size;
  const float* hs   = (const float*)d_in[0];
  const float* Wq   = (const float*)d_in[1];
  const float* bq   = (const float*)d_in[2];
  const float* Wk   = (const float*)d_in[3];
  const float* bk   = (const float*)d_in[4];
  const float* Wv   = (const float*)d_in[5];
  const float* bv   = (const float*)d_in[6];
  const float* Wo   = (const float*)d_in[7];
  const float* ssk  = (const float*)d_in[8];
  const float* ssv  = (const float*)d_in[9];
  const float* w0k  = (const float*)d_in[10];
  const float* w0v  = (const float*)d_in[11];
  const float* w1k  = (const float*)d_in[12];
  const float* w1v  = (const float*)d_in[13];
  const float* cosn = (const float*)d_in[14];
  const float* sinn = (const float*)d_in[15];
  float* ws  = (float*)d_ws;
  float* out = (float*)d_out;

  qkv_gemm<<<W_WORKERS * 384, 32, 0, stream>>>(hs, Wq, bq, Wk, bk, Wv, bv, ws);
  rope_k<<<(W_WORKERS * N_KV * Q_LEN * HEAD_DIM) / 256, 256, 0, stream>>>(cosn, sinn, ws);
  attn_partial<<<W_WORKERS * N_HEADS * CHUNKS, 32, 0, stream>>>(ssk, ssv, w0k, w0v, w1k, w1v, ws);
  attn_combine<<<W_WORKERS * N_HEADS, 256, 0, stream>>>(ws);
  out_gemm<<<W_WORKERS * 256, 32, 0, stream>>>(Wo, ws + WS_ATTN, out);
}